// LocationAttention_83485574300223
// MI455X (gfx1250) — compile-verified
//
#include <hip/hip_runtime.h>
#include <hip/hip_bf16.h>

// LocationAttention for MI455X (gfx1250, wave32).
// B=16, S=4096, H=512. out[b,t,h] = cumsum_s(a*x)/(cumsum_s(a)+eps),
// a = exp(x·w + bias). Memory-bound: ~400MB traffic -> ~17us @ 23.3TB/s.
// Causal chunk-scan via chained v_wmma_f32_16x16x4_f32:
//   Num(16x16) = L(16x16 lower-tri ones) * Y(16x16) + broadcast(carry)

#define B_    16
#define S_    4096
#define H_    512
#define NSEG  8
#define SEGLEN (S_ / NSEG)   // 512
#define NCH   (SEGLEN / 16)  // 32 chunks per segment
#define EPS_  1e-9f

typedef __attribute__((ext_vector_type(2))) float v2f;
typedef __attribute__((ext_vector_type(8))) float v8f;

// ---------------------------------------------------------------------------
// Kernel 1: scores a[b,s] = exp(<x[b,s,:],w> + bias), fused per-segment
// column sums segCol[b,g,h] = sum_{s in seg} a[s]*x[s,h].
// ---------------------------------------------------------------------------
__global__ __launch_bounds__(256) void k1_scores(
    const float* __restrict__ x, const float* __restrict__ w,
    const float* __restrict__ bias, float* __restrict__ aS,
    float* __restrict__ segCol)
{
  __shared__ float shCol[H_];
  const int blk  = blockIdx.x;           // b*NSEG + g
  const int b    = blk / NSEG;
  const int g    = blk % NSEG;
  const int tid  = threadIdx.x;
  const int wave = tid >> 5;
  const int lane = tid & 31;

  for (int i = tid; i < H_; i += 256) shCol[i] = 0.0f;
  __syncthreads();

  const float bval = bias[0];

  float4 wv[4];
  const float4* w4 = (const float4*)w;
  #pragma unroll
  for (int k = 0; k < 4; ++k) wv[k] = w4[lane + 32 * k];

  float4 colAcc[4];
  #pragma unroll
  for (int k = 0; k < 4; ++k) colAcc[k] = make_float4(0.f, 0.f, 0.f, 0.f);

  const int segRowBase = b * S_ + g * SEGLEN;
  for (int j = 0; j < SEGLEN / 8; ++j) {
    const int row = segRowBase + j * 8 + wave;       // 8 waves -> 8 consecutive rows
    const float4* xr = (const float4*)(x + (size_t)row * H_);
    float4 xv[4];
    float dot = 0.0f;
    #pragma unroll
    for (int k = 0; k < 4; ++k) {
      xv[k] = xr[lane + 32 * k];
      dot += xv[k].x * wv[k].x + xv[k].y * wv[k].y +
             xv[k].z * wv[k].z + xv[k].w * wv[k].w;
    }
    #pragma unroll
    for (int o = 16; o >= 1; o >>= 1) dot += __shfl_xor(dot, o, 32);
    const float a = __expf(dot + bval);
    if (lane == 0) aS[row] = a;
    #pragma unroll
    for (int k = 0; k < 4; ++k) {
      colAcc[k].x += a * xv[k].x; colAcc[k].y += a * xv[k].y;
      colAcc[k].z += a * xv[k].z; colAcc[k].w += a * xv[k].w;
    }
  }

  #pragma unroll
  for (int k = 0; k < 4; ++k) {
    const int h0 = 4 * (lane + 32 * k);
    atomicAdd(&shCol[h0 + 0], colAcc[k].x);
    atomicAdd(&shCol[h0 + 1], colAcc[k].y);
    atomicAdd(&shCol[h0 + 2], colAcc[k].z);
    atomicAdd(&shCol[h0 + 3], colAcc[k].w);
  }
  __syncthreads();
  float* dst = segCol + blk * H_;
  for (int i = tid; i < H_; i += 256) dst[i] = shCol[i];
}

// ---------------------------------------------------------------------------
// Kernel 2 (tiny): per batch — (A) den = cumsum(a[b,:]) block scan;
// (B) exclusive prefix of segCol across segments -> carry[b,g,h].
// ---------------------------------------------------------------------------
__global__ __launch_bounds__(512) void k2_prefix(
    const float* __restrict__ aS, const float* __restrict__ segCol,
    float* __restrict__ den, float* __restrict__ carry)
{
  __shared__ float sh[512];
  const int b = blockIdx.x, tid = threadIdx.x;

  const int base = b * S_ + tid * 8;
  float v[8];
  float run = 0.0f;
  #pragma unroll
  for (int j = 0; j < 8; ++j) { run += aS[base + j]; v[j] = run; }
  sh[tid] = run;
  __syncthreads();
  const float total = run;
  for (int off = 1; off < 512; off <<= 1) {
    const float t = (tid >= off) ? sh[tid - off] : 0.0f;
    __syncthreads();
    sh[tid] += t;
    __syncthreads();
  }
  const float excl = sh[tid] - total;
  #pragma unroll
  for (int j = 0; j < 8; ++j) den[base + j] = excl + v[j];

  const int h = tid;  // 512 threads == H_
  float run2 = 0.0f;
  for (int g = 0; g < NSEG; ++g) {
    const int idx = (b * NSEG + g) * H_ + h;
    carry[idx] = run2;
    run2 += segCol[idx];
  }
}

// ---------------------------------------------------------------------------
// Kernel 3: causal scan + normalize. One wave per (b, seg, 16-wide h-tile).
// Register double-buffering: chunk ch+1's 13 operand loads are issued before
// chunk ch's wmma chain, hiding memory latency under the serial carry dep.
// ---------------------------------------------------------------------------
__global__ __launch_bounds__(128) void k3_scan(
    const float* __restrict__ x, const float* __restrict__ aS,
    const float* __restrict__ den, const float* __restrict__ carry,
    float* __restrict__ out)
{
  const int wid  = blockIdx.x * 4 + (threadIdx.x >> 5); // global wave task
  const int ht   = wid & 31;            // H_/16 = 32 h-tiles
  const int bg   = wid >> 5;            // b*NSEG + g
  const int lane = threadIdx.x & 31;
  const int n    = lane & 15;           // column within tile (B/C/D layout)
  const int kb   = (lane < 16) ? 0 : 2; // K-half for A and B operands
  const int half = (lane < 16) ? 0 : 8; // M-half for C/D rows
  const int hbase   = ht * 16;
  const int g       = bg & (NSEG - 1);
  const int b       = bg >> 3;
  const int rowBase = b * S_ + g * SEGLEN;

  // Loop-invariant triangular A slabs: A_k[m, 4k+kb+c] = (4k+kb+c <= m)
  v2f A[4];
  #pragma unroll
  for (int k = 0; k < 4; ++k) {
    A[k].x = (4 * k + kb + 0 <= n) ? 1.0f : 0.0f;
    A[k].y = (4 * k + kb + 1 <= n) ? 1.0f : 0.0f;
  }

  float c = carry[bg * H_ + hbase + n];

  // double-buffered chunk operands (registers only)
  float  xb[2][8];
  float2 ab[2][4];
  float  db[2];

  auto loadChunk = [&](int buf, int r0_) {
    const float*  px = x + (size_t)(r0_ + kb) * H_ + hbase + n;
    const float2* pa = (const float2*)(aS + r0_ + kb);
    #pragma unroll
    for (int k = 0; k < 4; ++k) {
      xb[buf][2 * k]     = px[(4 * k) * H_];      // row r0+kb+4k
      xb[buf][2 * k + 1] = px[(4 * k + 1) * H_];  // row r0+kb+4k+1
      ab[buf][k]         = pa[2 * k];             // a[r0+kb+4k], a[...+1]
    }
    db[buf] = den[r0_ + n];
  };

  loadChunk(0, rowBase);

  #pragma unroll 2
  for (int ch = 0; ch < NCH; ++ch) {
    const int cur = ch & 1;
    const int r0  = rowBase + ch * 16;
    if (ch + 1 < NCH) loadChunk(cur ^ 1, r0 + 16);   // prefetch next chunk

    v8f acc;
    #pragma unroll
    for (int j = 0; j < 8; ++j) acc[j] = c;          // C[t,h] = carry[h]

    #pragma unroll
    for (int k = 0; k < 4; ++k) {
      v2f Bv;
      Bv.x = ab[cur][k].x * xb[cur][2 * k];
      Bv.y = ab[cur][k].y * xb[cur][2 * k + 1];
      acc = __builtin_amdgcn_wmma_f32_16x16x4_f32(
          false, A[k], false, Bv, (short)0, acc, false, false);
    }

    // carry[h] for next chunk = D[15, h] -> VGPR7 of lane n+16
    c = __shfl(acc[7], n + 16, 32);

    // fast reciprocal of den (v_rcp_f32, ~1ulp — fine for normalization)
    const float rden = __builtin_amdgcn_rcpf(db[cur] + EPS_);

    float* po = out + (size_t)(r0 + half) * H_ + hbase + n;
    #pragma unroll
    for (int j = 0; j < 8; ++j) {
      const float rd = __shfl(rden, half + j, 32);   // den row M = half+j
      po[j * H_] = acc[j] * rd;
    }
  }
}

// ---------------------------------------------------------------------------
extern "C" void kernel_launch(void* const* d_in, const int* in_sizes, int n_in,
                              void* d_out, int out_size, void* d_ws, size_t ws_size,
                              hipStream_t stream) {
  const float* x    = (const float*)d_in[0];   // [B,S,H] fp32
  const float* w    = (const float*)d_in[1];   // [H]
  const float* bias = (const float*)d_in[2];   // scalar
  float* out = (float*)d_out;

  // workspace: 4 * 65536 floats = 1 MB
  float* ws     = (float*)d_ws;
  float* aS     = ws;                 // [B*S]      scores
  float* den    = ws + B_ * S_;       // [B*S]      cumsum(a)
  float* segCol = ws + 2 * B_ * S_;   // [B*NSEG*H] segment column sums
  float* carry  = ws + 3 * B_ * S_;   // [B*NSEG*H] exclusive prefixes

  hipLaunchKernelGGL(k1_scores, dim3(B_ * NSEG), dim3(256), 0, stream,
                     x, w, bias, aS, segCol);
  hipLaunchKernelGGL(k2_prefix, dim3(B_), dim3(512), 0, stream,
                     aS, segCol, den, carry);
  hipLaunchKernelGGL(k3_scan, dim3(B_ * NSEG * (H_ / 16) / 4), dim3(128), 0, stream,
                     x, aS, den, carry, out);
}